// BornFoward_51883204935991
// MI455X (gfx1250) — compile-verified
//
#include <hip/hip_runtime.h>

// ---------------------------------------------------------------------------
// Born forward wave solver for MI455X (gfx1250) — single-launch, LDS-resident.
//
// CDNA5 gives 320 KB LDS per workgroup: a whole 192x192 f32 plane ping-pong
// (2 x 192 x 196(pad) x 4 B = 301 KB) fits. One workgroup per (batch,source)
// plane runs all 200 timesteps with one s_barrier per step -> ONE kernel
// launch instead of 402 (graph-replay launch overhead ~0.4 ms vs the ~10 us
// HBM floor of streaming P0 once). Global traffic is only the P0 stream,
// tiny x reads, and 32 receiver samples per plane per step.
//
//   lap4(p) = L*p + p*L,  L = circulant([-1,16,-30,16,-1]/12) (symmetric)
// 192 = 12x16 blocking; band halfwidth 2 -> per side a sliding window of SIX
// nonzero 16x16x4 chunks (q=0..5, circulant K-range 16T-4+4q..+3, weight
// offset d0=4q-4). 12 x V_WMMA_F32_16X16X4_F32 per tile, no zero operands.
// Wave w of 32 handles tiles {w, w+32, ...} (wave-uniform -> EXEC all ones).
// ---------------------------------------------------------------------------

typedef __attribute__((ext_vector_type(2))) float v2f;
typedef __attribute__((ext_vector_type(8))) float v8f;

#define NXG   192       // padded grid
#define NTILE 12        // 16-wide blocks per axis
#define NT    200       // timesteps
#define NM    32        // receivers
#define PLSZ  (NXG*NXG)
#define LDSP  196       // padded LDS row stride (floats): bank step 4, 8B-aligned

// stencil weight w(d): w(0)=-30/12, w(+-1)=16/12, w(+-2)=-1/12, else 0
__device__ __forceinline__ float wfun(int d) {
    int ad = d < 0 ? -d : d;
    float v = 0.0f;
    v = (ad == 2) ? (-1.0f / 12.0f)  : v;
    v = (ad == 1) ? (16.0f / 12.0f)  : v;
    v = (ad == 0) ? (-30.0f / 12.0f) : v;
    return v;
}

__device__ __forceinline__ int wrap192(int r) {
    r += (r < 0)    ? NXG : 0;
    r -= (r >= NXG) ? NXG : 0;
    return r;
}

__global__ __launch_bounds__(1024) void born_fused(
    const float* __restrict__ x,    // (2,1,96,96)
    const float* __restrict__ P0,   // (1,8,200,192,192)
    float*       __restrict__ out)  // (2,8,32,200)
{
    // step j reads p1 = S[(j&1)^1], reads/writes (in place) p0/pnew = S[j&1]
    __shared__ float S[2][NXG * LDSP];            // 301,056 B < 320 KB

    const int tid   = threadIdx.x;
    const int lane  = tid & 31;
    const int wv    = tid >> 5;                   // wave id 0..31
    const int plane = blockIdx.x;                 // 0..15 = b*8 + nr
    const int b     = plane >> 3;
    const int nr    = plane & 7;

    // WMMA f32 lane mapping (wave32):
    //  A 16x4 : lane<16 -> M=lane, K in {0,1}; lane>=16 -> M=lane-16, K in {2,3}
    //  B 4x16 : same with N;  C/D: VGPR r -> (M = r + 8*(lane>=16), N = lane&15)
    const int half = lane >> 4;
    const int lm   = lane & 15;
    const int k0   = half * 2;

    // ---- zero both field buffers (fields start at rest)
    for (int i = tid; i < 2 * NXG * LDSP; i += 1024) ((float*)S)[i] = 0.0f;

    // ---- receiver coordinates, one per lane 0..31
    int mx = 0, my = 0;
    if (tid < NM) {
        const float th = 6.283185307179586f * (float)tid / 32.0f;
        mx = (int)(96.0f + 70.0f * __cosf(th));
        my = (int)(96.0f + 70.0f * __sinf(th));
    }

    // ---- hoisted circulant L operands: 6 v2f constants; chunk q multiplies
    // with weight w(4q-4 + k - m). Same register serves as the A operand of
    // the x-part and (L symmetric, w even) the B operand of the y-part.
    v2f Lw[6];
#pragma unroll
    for (int q = 0; q < 6; ++q) {
        const int d = 4 * q - 4 + k0 - lm;
        Lw[q].x = wfun(d);
        Lw[q].y = wfun(d + 1);
    }

    const float* P0n = P0 + (size_t)nr * NT * PLSZ;

    __syncthreads();

    for (int j = 0; j < NT; ++j) {
        float*       wb = S[j & 1];               // p0 in, pnew out (in place)
        const float* rb = S[(j & 1) ^ 1];         // p1

        // 144 tiles round-robin over 32 waves (wave-uniform loop bounds)
        for (int tl = wv; tl < NTILE * NTILE; tl += 32) {
            const int I = tl / NTILE;
            const int J = tl % NTILE;

            v8f acc = {0.f, 0.f, 0.f, 0.f, 0.f, 0.f, 0.f, 0.f};

            // x-part: (L*p1); B chunk rows = circulant rows 16I-4+4q+k0, +1
#pragma unroll
            for (int q = 0; q < 6; ++q) {
                const int rr = I * 16 - 4 + 4 * q + k0;
                const int r0 = wrap192(rr);
                const int r1 = wrap192(rr + 1);
                v2f bm;
                bm.x = rb[r0 * LDSP + J * 16 + lm];
                bm.y = rb[r1 * LDSP + J * 16 + lm];
                acc = __builtin_amdgcn_wmma_f32_16x16x4_f32(
                          false, Lw[q], false, bm, (short)0, acc, false, false);
            }

            // y-part: (p1*L); A chunk = row 16I+lm, circulant cols
            // 16J-4+4q+k0 (even, pair never straddles the wrap) -> b64 load
#pragma unroll
            for (int q = 0; q < 6; ++q) {
                const int col = wrap192(J * 16 - 4 + 4 * q + k0);
                const v2f a = *(const v2f*)&rb[(I * 16 + lm) * LDSP + col];
                acc = __builtin_amdgcn_wmma_f32_16x16x4_f32(
                          false, a, false, Lw[q], (short)0, acc, false, false);
            }

            // elementwise epilogue (after all WMMAs of this tile)
#pragma unroll
            for (int r = 0; r < 8; ++r) {
                const int row = I * 16 + half * 8 + r;
                const int col = J * 16 + lm;
                const int li  = row * LDSP + col;

                const float p1v = rb[li];
                const float p0v = wb[li];

                // BGF: 1.5 inside the 25-wide ABC ring, 1e-6 outside
                const bool  inb  = (row >= 25) & (row < 167) & (col >= 25) & (col < 167);
                const float bgf  = inb ? 1.5f : 1e-6f;
                const float coef = (0.3f * bgf) * (0.3f * bgf);   // (dt*BGF/dx)^2

                // X = pad(bg/x, 48 each side, value 1)
                float Xv = 1.0f;
                if ((row >= 48) & (row < 144) & (col >= 48) & (col < 144)) {
                    Xv = 1.5f / x[(size_t)(b * 96 + (row - 48)) * 96 + (col - 48)];
                }
                const float rf = (bgf * bgf) * (1.0f / 2.25f) * (1.0f - Xv * Xv);

                // d2 = P0[j] - 2*P0[j-1] + P0[j-2]; zero for j<2
                float d2v = 0.0f;
                if (j >= 2) {
                    const size_t pi = (size_t)j * PLSZ + (size_t)row * NXG + col;
                    d2v = P0n[pi] - 2.0f * P0n[pi - PLSZ] + P0n[pi - 2 * (size_t)PLSZ];
                }

                wb[li] = 2.0f * p1v - p0v + coef * acc[r] + rf * d2v;
            }
        }

        // pull next timestep's P0 slice toward L2 (global_prefetch_b8)
        if (j + 1 < NT) {
            __builtin_prefetch(P0n + (size_t)(j + 1) * PLSZ + (size_t)tid * 36, 0, 1);
        }

        __syncthreads();   // pnew complete everywhere before sampling / next step

        // receiver sampling straight from LDS
        if (tid < NM) {
            out[((size_t)plane * NM + tid) * NT + j] = wb[mx * LDSP + my];
        }
    }
}

extern "C" void kernel_launch(void* const* d_in, const int* in_sizes, int n_in,
                              void* d_out, int out_size, void* d_ws, size_t ws_size,
                              hipStream_t stream) {
    (void)in_sizes; (void)n_in; (void)d_ws; (void)ws_size; (void)out_size;
    const float* x  = (const float*)d_in[0];   // (2,1,96,96)  f32
    const float* P0 = (const float*)d_in[1];   // (1,8,200,192,192) f32
    float* out = (float*)d_out;                // (2,8,32,200) f32

    // One block per (batch,source) plane; entire time loop inside the kernel.
    born_fused<<<16, 1024, 0, stream>>>(x, P0, out);
}